// FlexNet_57810259804754
// MI455X (gfx1250) — compile-verified
//
#include <hip/hip_runtime.h>
#include <math.h>

// ============================================================================
// FlexNet GNN forward for MI455X (gfx1250, wave32).
// Bandwidth-bound graph net: scalar/VALU kernels for the tiny-channel
// gather/scatter message passing (L2-resident node tables, f32 atomics for
// segment_sum, f64 atomics for BN statistics), and V_WMMA_F32_16X16X4_F32
// for the one dense GEMM of useful shape: edge-MLP [E,28] @ [28,14].
// ============================================================================

typedef float v2f __attribute__((ext_vector_type(2)));
typedef float v8f __attribute__((ext_vector_type(8)));

#define ATOM_ADD(p, v) __hip_atomic_fetch_add((p), (v), __ATOMIC_RELAXED, __HIP_MEMORY_SCOPE_AGENT)

__device__ __forceinline__ float sigmoid15(float z) { return 1.f / (1.f + expf(-1.5f * z)); }

// ---------------------------------------------------------------------------
// small utility kernels
// ---------------------------------------------------------------------------
__global__ __launch_bounds__(256) void zero_stats_k(double* a, double* b) {
  int i = threadIdx.x;
  if (i < 16) { a[i] = 0.0; b[i] = 0.0; }
}

// out[:,0] = x ; out[:,1..] = 0   (column 0 of every layer output is x)
__global__ __launch_bounds__(256) void init_out_k(float* __restrict__ o,
                                                  const float* __restrict__ x,
                                                  int n_, int width) {
  int gid = blockIdx.x * blockDim.x + threadIdx.x;
  int stride = gridDim.x * blockDim.x;
  for (int n = gid; n < n_; n += stride) {
    float* row = o + (size_t)n * width;
    row[0] = x[n];
    for (int j = 1; j < width; j++) row[j] = 0.f;
  }
}

// scale = g * rsqrt(var+eps) ; shift = b - mu*scale  (biased variance)
__global__ __launch_bounds__(256) void bn_finalize_k(const double* __restrict__ ds,
                                                     const double* __restrict__ dq,
                                                     const float* __restrict__ g,
                                                     const float* __restrict__ b,
                                                     double cntInv, int co,
                                                     float* __restrict__ scale,
                                                     float* __restrict__ shift) {
  int c = threadIdx.x;
  if (c < co) {
    double mu  = ds[c] * cntInv;
    double var = dq[c] * cntInv - mu * mu;
    float s = g[c] / sqrtf((float)var + 1e-5f);
    scale[c] = s;
    shift[c] = b[c] - (float)mu * s;
  }
}

// ---------------------------------------------------------------------------
// message-passing layers (flexconv: EA=true, edgeinter: EA=false)
// t[c] = h[dst]·Ws[:,c] + h[src]·Wn[:,c] (+ ea·We[:,c])
// pass1: channel-wise sum/sumsq over E (wave shfl -> LDS -> f64 atomics)
// pass2: m = relu(scale*t + shift); atomic scatter-add into out[dst, 1+c]
// ---------------------------------------------------------------------------
template <int CI, int CO, bool EA>
__device__ __forceinline__ void conv_msg(const float* __restrict__ h,
                                         const float* __restrict__ ea,
                                         const float* sWs, const float* sWn, const float* sWe,
                                         int e, int s, int d, float (&t)[CO]) {
  float hd[CI], hs[CI];
#pragma unroll
  for (int k = 0; k < CI; k++) { hd[k] = h[(size_t)d * CI + k]; hs[k] = h[(size_t)s * CI + k]; }
  float e0 = 0.f, e1 = 0.f;
  if (EA) { e0 = ea[(size_t)e * 2]; e1 = ea[(size_t)e * 2 + 1]; }
#pragma unroll
  for (int c = 0; c < CO; c++) {
    float acc = 0.f;
#pragma unroll
    for (int k = 0; k < CI; k++) acc += hd[k] * sWs[k * CO + c] + hs[k] * sWn[k * CO + c];
    if (EA) acc += e0 * sWe[c] + e1 * sWe[CO + c];
    t[c] = acc;
  }
}

template <int CI, int CO, bool EA>
__global__ __launch_bounds__(256) void conv_pass1(const float* __restrict__ h,
                                                  const int* __restrict__ eidx, int E_,
                                                  const float* __restrict__ ea,
                                                  const float* __restrict__ Ws,
                                                  const float* __restrict__ Wn,
                                                  const float* __restrict__ We,
                                                  double* __restrict__ dsum,
                                                  double* __restrict__ dsq) {
  __shared__ float sWs[CI * CO], sWn[CI * CO];
  __shared__ float sWe[EA ? 2 * CO : 2];
  __shared__ float sPs[8][CO], sPq[8][CO];
  int tid = threadIdx.x;
  for (int i = tid; i < CI * CO; i += blockDim.x) { sWs[i] = Ws[i]; sWn[i] = Wn[i]; }
  if (EA) for (int i = tid; i < 2 * CO; i += blockDim.x) sWe[i] = We[i];
  __syncthreads();

  float ls[CO], lq[CO];
#pragma unroll
  for (int c = 0; c < CO; c++) { ls[c] = 0.f; lq[c] = 0.f; }

  int gid = blockIdx.x * blockDim.x + tid;
  int stride = gridDim.x * blockDim.x;
  for (int e = gid; e < E_; e += stride) {
    int s = eidx[e], d = eidx[E_ + e];
    float t[CO];
    conv_msg<CI, CO, EA>(h, ea, sWs, sWn, sWe, e, s, d, t);
#pragma unroll
    for (int c = 0; c < CO; c++) { ls[c] += t[c]; lq[c] += t[c] * t[c]; }
  }

  // wave32 tree reduce, then per-block f64 accumulation
#pragma unroll
  for (int c = 0; c < CO; c++) {
    float v = ls[c], q = lq[c];
    for (int m = 16; m > 0; m >>= 1) { v += __shfl_xor(v, m, 32); q += __shfl_xor(q, m, 32); }
    ls[c] = v; lq[c] = q;
  }
  int wave = tid >> 5, lane = tid & 31;
  if (lane == 0) {
#pragma unroll
    for (int c = 0; c < CO; c++) { sPs[wave][c] = ls[c]; sPq[wave][c] = lq[c]; }
  }
  __syncthreads();
  if (tid < CO) {
    double a = 0.0, b = 0.0;
    for (int w = 0; w < 8; w++) { a += (double)sPs[w][tid]; b += (double)sPq[w][tid]; }
    ATOM_ADD(&dsum[tid], a);
    ATOM_ADD(&dsq[tid], b);
  }
}

template <int CI, int CO, bool EA>
__global__ __launch_bounds__(256) void conv_pass2(const float* __restrict__ h,
                                                  const int* __restrict__ eidx, int E_,
                                                  const float* __restrict__ ea,
                                                  const float* __restrict__ Ws,
                                                  const float* __restrict__ Wn,
                                                  const float* __restrict__ We,
                                                  const float* __restrict__ scale,
                                                  const float* __restrict__ shift,
                                                  float* __restrict__ outp) {
  __shared__ float sWs[CI * CO], sWn[CI * CO];
  __shared__ float sWe[EA ? 2 * CO : 2];
  __shared__ float sSc[CO], sSh[CO];
  int tid = threadIdx.x;
  for (int i = tid; i < CI * CO; i += blockDim.x) { sWs[i] = Ws[i]; sWn[i] = Wn[i]; }
  if (EA) for (int i = tid; i < 2 * CO; i += blockDim.x) sWe[i] = We[i];
  if (tid < CO) { sSc[tid] = scale[tid]; sSh[tid] = shift[tid]; }
  __syncthreads();

  int gid = blockIdx.x * blockDim.x + tid;
  int stride = gridDim.x * blockDim.x;
  for (int e = gid; e < E_; e += stride) {
    int s = eidx[e], d = eidx[E_ + e];
    float t[CO];
    conv_msg<CI, CO, EA>(h, ea, sWs, sWn, sWe, e, s, d, t);
#pragma unroll
    for (int c = 0; c < CO; c++) {
      float m = sSc[c] * t[c] + sSh[c];
      m = m > 0.f ? m : 0.f;
      if (m != 0.f) ATOM_ADD(&outp[(size_t)d * (CO + 1) + 1 + c], m);  // relu-zeros need no atomic
    }
  }
}

// ---------------------------------------------------------------------------
// node MLP layer: r = relu((AFF ? in*sc+sh : in) @ W + bias), store r + stats
// ---------------------------------------------------------------------------
template <int CI, int CO, bool AFF>
__global__ __launch_bounds__(256) void mlp_pass(const float* __restrict__ hin, int n_,
                                                const float* __restrict__ inSc,
                                                const float* __restrict__ inSh,
                                                const float* __restrict__ W,
                                                const float* __restrict__ bias,
                                                float* __restrict__ rout,
                                                double* __restrict__ dsum,
                                                double* __restrict__ dsq) {
  __shared__ float sW[CI * CO], sB[CO];
  __shared__ float sSc[AFF ? CI : 1], sSh[AFF ? CI : 1];
  __shared__ float sPs[8][CO], sPq[8][CO];
  int tid = threadIdx.x;
  for (int i = tid; i < CI * CO; i += blockDim.x) sW[i] = W[i];
  if (tid < CO) sB[tid] = bias[tid];
  if (AFF && tid < CI) { sSc[tid] = inSc[tid]; sSh[tid] = inSh[tid]; }
  __syncthreads();

  float ls[CO], lq[CO];
#pragma unroll
  for (int c = 0; c < CO; c++) { ls[c] = 0.f; lq[c] = 0.f; }

  int gid = blockIdx.x * blockDim.x + tid;
  int stride = gridDim.x * blockDim.x;
  for (int n = gid; n < n_; n += stride) {
    float hv[CI];
#pragma unroll
    for (int k = 0; k < CI; k++) {
      float v = hin[(size_t)n * CI + k];
      if (AFF) v = sSc[k] * v + sSh[k];
      hv[k] = v;
    }
#pragma unroll
    for (int c = 0; c < CO; c++) {
      float r = sB[c];
#pragma unroll
      for (int k = 0; k < CI; k++) r += hv[k] * sW[k * CO + c];
      r = r > 0.f ? r : 0.f;
      rout[(size_t)n * CO + c] = r;
      ls[c] += r; lq[c] += r * r;
    }
  }

#pragma unroll
  for (int c = 0; c < CO; c++) {
    float v = ls[c], q = lq[c];
    for (int m = 16; m > 0; m >>= 1) { v += __shfl_xor(v, m, 32); q += __shfl_xor(q, m, 32); }
    ls[c] = v; lq[c] = q;
  }
  int wave = tid >> 5, lane = tid & 31;
  if (lane == 0) {
#pragma unroll
    for (int c = 0; c < CO; c++) { sPs[wave][c] = ls[c]; sPq[wave][c] = lq[c]; }
  }
  __syncthreads();
  if (tid < CO) {
    double a = 0.0, b = 0.0;
    for (int w = 0; w < 8; w++) { a += (double)sPs[w][tid]; b += (double)sPq[w][tid]; }
    ATOM_ADD(&dsum[tid], a);
    ATOM_ADD(&dsq[tid], b);
  }
}

// power = sigmoid(1.5 * ((sc*r2+sh) @ pw3 + pb3))
__global__ __launch_bounds__(256) void power_k(const float* __restrict__ r2, int n_,
                                               const float* __restrict__ sc,
                                               const float* __restrict__ sh,
                                               const float* __restrict__ pw3,
                                               const float* __restrict__ pb3,
                                               float* __restrict__ outPower) {
  int gid = blockIdx.x * blockDim.x + threadIdx.x;
  int stride = gridDim.x * blockDim.x;
  float w0 = pw3[0], w1 = pw3[1], w2 = pw3[2], w3 = pw3[3];
  float s0 = sc[0], s1 = sc[1], s2 = sc[2], s3 = sc[3];
  float h0 = sh[0], h1 = sh[1], h2 = sh[2], h3 = sh[3];
  float b0 = pb3[0];
  for (int n = gid; n < n_; n += stride) {
    const float* r = r2 + (size_t)n * 4;
    float z = b0 + (s0 * r[0] + h0) * w0 + (s1 * r[1] + h1) * w1 +
                    (s2 * r[2] + h2) * w2 + (s3 * r[3] + h3) * w3;
    outPower[n] = sigmoid15(z);
  }
}

// ---------------------------------------------------------------------------
// edge MLP via V_WMMA_F32_16X16X4_F32.
// f[e] = [out13[ddst](13), power[ddst], out13[dsrc](13), power[dsrc]]  (K=28)
// One wave = one 16-edge tile; 7 WMMAs (K chunks of 4), N=14 padded to 16.
// ISA layouts (f32): A lane L -> row L%16, ks {2*half, 2*half+1};
//                    B VGPR0 -> K rows {2*half..}, cols L%16;
//                    D VGPRv at lane L -> row v+8*half, col L%16.
// Tail tiles: index is CLAMPED (always load real data, no exec-masked loads);
// garbage rows are zero-selected out of the stats / skipped on store.
// ---------------------------------------------------------------------------
__device__ __forceinline__ float f_elem(const float* __restrict__ o13,
                                        const float* __restrict__ pw,
                                        int dd, int ds, int k) {
  // branch-free: two selects on the address, single load
  int node = (k < 14) ? dd : ds;
  int kk   = (k < 14) ? k : (k - 14);
  const float* p = (kk < 13) ? (o13 + (size_t)node * 13 + kk) : (pw + node);
  return *p;
}

template <int PASS>
__global__ __launch_bounds__(256) void emlp_k(const float* __restrict__ o13,
                                              const float* __restrict__ pw,
                                              const int* __restrict__ didx, int E_,
                                              const float* __restrict__ ew1,
                                              const float* __restrict__ eb1,
                                              double* __restrict__ dsum,
                                              double* __restrict__ dsq,
                                              const float* __restrict__ scale,
                                              const float* __restrict__ shift,
                                              const float* __restrict__ ew2,
                                              const float* __restrict__ eb2,
                                              float* __restrict__ dirOut) {
  int tid = blockIdx.x * blockDim.x + threadIdx.x;
  int lane = threadIdx.x & 31;
  int n = lane & 15;        // output channel (14 valid, 2 pad)
  int half = lane >> 4;     // K/row half-select per ISA layout
  int waveId = tid >> 5;
  int numWaves = (gridDim.x * blockDim.x) >> 5;
  bool chanOk = (n < 14);

  // B operand: ew1 [28,14] row-major, K chunks of 4, columns padded to 16.
  v2f B[7];
#pragma unroll
  for (int c = 0; c < 7; c++) {
    int k0 = 4 * c + 2 * half;
    B[c].x = chanOk ? ew1[k0 * 14 + n] : 0.f;
    B[c].y = chanOk ? ew1[(k0 + 1) * 14 + n] : 0.f;
  }
  float bias = chanOk ? eb1[n] : 0.f;
  float sc = 0.f, sf = 0.f, w2 = 0.f, b2 = 0.f;
  if (PASS == 2) {
    if (chanOk) { sc = scale[n]; sf = shift[n]; w2 = ew2[n]; }
    b2 = eb2[0];
  }

  double lsum = 0.0, lsq = 0.0;
  int tiles = (E_ + 15) >> 4;
  for (int t = waveId; t < tiles; t += numWaves) {
    int base = t << 4;
    int e = base + n;                       // A row this lane gathers
    int ec = (e < E_) ? e : (E_ - 1);       // clamp: unconditional loads
    int ds_ = didx[ec];
    int dd_ = didx[E_ + ec];

    // issue all 14 A-element gathers up front so VMEM latency overlaps,
    // then run the uninterrupted 7-WMMA accumulation chain
    float av[14];
#pragma unroll
    for (int c = 0; c < 7; c++) {
      int k0 = 4 * c + 2 * half;
      av[2 * c]     = f_elem(o13, pw, dd_, ds_, k0);
      av[2 * c + 1] = f_elem(o13, pw, dd_, ds_, k0 + 1);
    }

    v8f acc = {0.f, 0.f, 0.f, 0.f, 0.f, 0.f, 0.f, 0.f};
#pragma unroll
    for (int c = 0; c < 7; c++) {
      v2f a;
      a.x = av[2 * c];
      a.y = av[2 * c + 1];
      acc = __builtin_amdgcn_wmma_f32_16x16x4_f32(false, a, false, B[c],
                                                  (short)0, acc, false, false);
    }

    if (PASS == 1) {
      // branchless stats: select invalid rows / pad channels to 0
#pragma unroll
      for (int v = 0; v < 8; v++) {
        int ed = base + v + 8 * half;
        float r = acc[v] + bias;
        r = r > 0.f ? r : 0.f;
        r = (ed < E_ && chanOk) ? r : 0.f;
        double rd = (double)r;
        lsum += rd;
        lsq += rd * rd;
      }
    } else {
#pragma unroll
      for (int v = 0; v < 8; v++) {
        float r = acc[v] + bias;
        r = r > 0.f ? r : 0.f;
        float p = chanOk ? (sc * r + sf) * w2 : 0.f;
        p += __shfl_xor(p, 1, 32);
        p += __shfl_xor(p, 2, 32);
        p += __shfl_xor(p, 4, 32);
        p += __shfl_xor(p, 8, 32);     // sum over the 16 channel lanes in this half
        int ed = base + v + 8 * half;
        if (n == 0 && ed < E_) dirOut[ed] = sigmoid15(p + b2);
      }
    }
  }

  if (PASS == 1) {
    lsum += __shfl_xor(lsum, 16, 32);  // combine the two halves (same channel)
    lsq  += __shfl_xor(lsq, 16, 32);
    if (half == 0 && chanOk) {
      ATOM_ADD(&dsum[n], lsum);
      ATOM_ADD(&dsq[n], lsq);
    }
  }
}

// ---------------------------------------------------------------------------
// host orchestration (all launches on `stream`, graph-capture safe)
// ---------------------------------------------------------------------------
extern "C" void kernel_launch(void* const* d_in, const int* in_sizes, int n_in,
                              void* d_out, int out_size, void* d_ws, size_t ws_size,
                              hipStream_t stream) {
  const float* x  = (const float*)d_in[0];
  const float* ea = (const float*)d_in[1];
  const int* ei = (const int*)d_in[2];   // [2,E]: src row0, dst row1
  const int* di = (const int*)d_in[3];   // dir edges
  const int* pi = (const int*)d_in[4];   // prop edges
#define F(i) ((const float*)d_in[(i)])
  const float *c1_ws = F(5), *c1_wn = F(6), *c1_we = F(7), *c1_g = F(8), *c1_b = F(9);
  const float *d1_ws = F(10), *d1_wn = F(11), *d1_g = F(12), *d1_b = F(13);
  const float *c2_ws = F(14), *c2_wn = F(15), *c2_we = F(16), *c2_g = F(17), *c2_b = F(18);
  const float *d2_ws = F(19), *d2_wn = F(20), *d2_g = F(21), *d2_b = F(22);
  const float *c3_ws = F(23), *c3_wn = F(24), *c3_we = F(25), *c3_g = F(26), *c3_b = F(27);
  const float *d3_ws = F(28), *d3_wn = F(29), *d3_g = F(30), *d3_b = F(31);
  const float *pw1 = F(32), *pb1 = F(33), *p1_g = F(34), *p1_b = F(35);
  const float *pw2 = F(36), *pb2 = F(37), *p2_g = F(38), *p2_b = F(39);
  const float *pw3 = F(40), *pb3 = F(41);
  const float *ew1 = F(42), *eb1 = F(43), *e1_g = F(44), *e1_b = F(45);
  const float *ew2 = F(46), *eb2 = F(47);
#undef F

  const int N = in_sizes[0];
  const int E = in_sizes[1] / 2;
  float* out = (float*)d_out;           // [0,N): power ; [N,N+E): direction

  // workspace carving (~15.2 MB)
  double* dsum = (double*)d_ws;
  double* dsq  = dsum + 16;
  float* scA = (float*)(dsq + 16);
  float* shA = scA + 16;
  float* r1  = shA + 16;                // [N,8]
  float* r2  = r1 + (size_t)N * 8;      // [N,4]
  float* bufA = r2 + (size_t)N * 4;     // [N,13] max
  float* bufB = bufA + (size_t)N * 13;  // [N,13] max

  const int TB = 256;
  const int NB_STAT = 2048, NB_SCAT = 4096, NB_NODE = 1024, NB_EMLP = 512;

#define CONV_LAYER(CI, CO, HASEA, HIN, HOUT, WS, WN, WE, Gp, Bp, EIDX, EAP)                        \
  zero_stats_k<<<1, 32, 0, stream>>>(dsum, dsq);                                                   \
  init_out_k<<<NB_NODE, TB, 0, stream>>>(HOUT, x, N, (CO) + 1);                                    \
  conv_pass1<CI, CO, HASEA><<<NB_STAT, TB, 0, stream>>>(HIN, EIDX, E, EAP, WS, WN, WE, dsum, dsq); \
  bn_finalize_k<<<1, 32, 0, stream>>>(dsum, dsq, Gp, Bp, 1.0 / (double)E, CO, scA, shA);           \
  conv_pass2<CI, CO, HASEA><<<NB_SCAT, TB, 0, stream>>>(HIN, EIDX, E, EAP, WS, WN, WE, scA, shA, HOUT);

  // 6 message-passing layers (col 0 of every output is x)
  CONV_LAYER(1, 2, true,  x,    bufA, c1_ws, c1_wn, c1_we, c1_g, c1_b, ei, ea)       // [N,3]
  CONV_LAYER(3, 4, false, bufA, bufB, d1_ws, d1_wn, (const float*)nullptr, d1_g, d1_b, pi, (const float*)nullptr) // [N,5]
  CONV_LAYER(5, 6, true,  bufB, bufA, c2_ws, c2_wn, c2_we, c2_g, c2_b, ei, ea)       // [N,7]
  CONV_LAYER(7, 8, false, bufA, bufB, d2_ws, d2_wn, (const float*)nullptr, d2_g, d2_b, pi, (const float*)nullptr) // [N,9]
  CONV_LAYER(9, 10, true, bufB, bufA, c3_ws, c3_wn, c3_we, c3_g, c3_b, ei, ea)       // [N,11]
  CONV_LAYER(11, 12, false, bufA, bufB, d3_ws, d3_wn, (const float*)nullptr, d3_g, d3_b, pi, (const float*)nullptr) // [N,13] -> out13 = bufB
#undef CONV_LAYER

  // node MLP: h1 = BN(relu(out13@pw1+pb1)); h2 = BN(relu(h1@pw2+pb2)); power
  zero_stats_k<<<1, 32, 0, stream>>>(dsum, dsq);
  mlp_pass<13, 8, false><<<NB_STAT, TB, 0, stream>>>(bufB, N, nullptr, nullptr, pw1, pb1, r1, dsum, dsq);
  bn_finalize_k<<<1, 32, 0, stream>>>(dsum, dsq, p1_g, p1_b, 1.0 / (double)N, 8, scA, shA);
  zero_stats_k<<<1, 32, 0, stream>>>(dsum, dsq);
  mlp_pass<8, 4, true><<<NB_STAT, TB, 0, stream>>>(r1, N, scA, shA, pw2, pb2, r2, dsum, dsq);
  bn_finalize_k<<<1, 32, 0, stream>>>(dsum, dsq, p2_g, p2_b, 1.0 / (double)N, 4, scA, shA);
  power_k<<<NB_NODE, TB, 0, stream>>>(r2, N, scA, shA, pw3, pb3, out);

  // edge MLP (WMMA): e1 = BN(relu(f@ew1+eb1)); direction = sigmoid(1.5*(e1@ew2+eb2))
  zero_stats_k<<<1, 32, 0, stream>>>(dsum, dsq);
  emlp_k<1><<<NB_EMLP, TB, 0, stream>>>(bufB, out, di, E, ew1, eb1, dsum, dsq,
                                        nullptr, nullptr, nullptr, nullptr, nullptr);
  bn_finalize_k<<<1, 32, 0, stream>>>(dsum, dsq, e1_g, e1_b, 1.0 / (double)E, 14, scA, shA);
  emlp_k<2><<<NB_EMLP, TB, 0, stream>>>(bufB, out, di, E, ew1, eb1, nullptr, nullptr,
                                        scA, shA, ew2, eb2, out + N);
}